// GCN_4758823764089
// MI455X (gfx1250) — compile-verified
//
#include <hip/hip_runtime.h>
#include <hip/hip_bf16.h>
#include <stdint.h>

typedef __attribute__((ext_vector_type(2))) float v2f;
typedef __attribute__((ext_vector_type(8))) float v8f;

#define N_NODES 75000
#define N_EDGES 1200000

// ---------------------------------------------------------------------------
// Degree / normalization kernels
// ---------------------------------------------------------------------------
__global__ void gcn_deg_init(float* __restrict__ deg, int n) {
    int i = blockIdx.x * blockDim.x + threadIdx.x;
    if (i < n) deg[i] = 1.0f;  // self-loop contributes 1 to every node's degree
}

__global__ void gcn_deg_count(const int* __restrict__ dst, float* __restrict__ deg, int nE) {
    int e = blockIdx.x * blockDim.x + threadIdx.x;
    if (e < nE) atomicAdd(&deg[dst[e]], 1.0f);
}

__global__ void gcn_deg_rsqrt(float* __restrict__ deg, int n) {
    int i = blockIdx.x * blockDim.x + threadIdx.x;
    if (i < n) deg[i] = rsqrtf(deg[i]);  // deg >= 1 always (self loops)
}

// ---------------------------------------------------------------------------
// WMMA GEMM: C[M x NDIM] = A[M x KDIM] * W[KDIM x NDIM], full fp32 via
// V_WMMA_F32_16X16X4_F32. One wave -> one 16x16 output tile. Block holds
// NDIM/16 waves covering the N dimension; grid.x covers M tiles.
//
// A operand layout (32-bit A 16x4): lanes 0-15 hold row M=lane with K={k,k+1}
// in the float2; lanes 16-31 hold row M=lane-16 with K={k+2,k+3}.
// B operand layout: lane holds column N=lane%16; lanes 0-15 carry K={k,k+1},
// lanes 16-31 carry K={k+2,k+3}.
// C/D layout: VGPR r -> M = r + (lane<16 ? 0 : 8), N = lane%16.
// ---------------------------------------------------------------------------
template <int KDIM, int NDIM>
__global__ void gcn_wmma_gemm(const float* __restrict__ A, const float* __restrict__ W,
                              float* __restrict__ C, int M) {
    const int lane  = threadIdx.x & 31;
    const int wave  = threadIdx.x >> 5;
    const int half  = lane >> 4;   // 0: lanes 0-15, 1: lanes 16-31
    const int l     = lane & 15;
    const int ntile = wave * 16;
    const int mtile = blockIdx.x * 16;

    int row = mtile + l;
    if (row >= M) row = M - 1;            // branchless clamp: EXEC stays all-1s
    const float* arow = A + (size_t)row * KDIM;
    const int n = ntile + l;              // this lane's output column

    v8f c = {};
#pragma unroll
    for (int k0 = 0; k0 < KDIM; k0 += 4) {
        const int ka = k0 + half * 2;
        v2f a, b;
        a.x = arow[ka];
        a.y = arow[ka + 1];
        b.x = W[(size_t)ka * NDIM + n];
        b.y = W[(size_t)(ka + 1) * NDIM + n];
        c = __builtin_amdgcn_wmma_f32_16x16x4_f32(false, a, false, b,
                                                  (short)0, c, false, false);
    }

#pragma unroll
    for (int r = 0; r < 8; ++r) {
        const int m = mtile + r + half * 8;
        if (m < M) C[(size_t)m * NDIM + n] = c[r];
    }
}

// ---------------------------------------------------------------------------
// Aggregation kernels
// ---------------------------------------------------------------------------
// out[i,f] = h[i,f] * dinv[i]^2   (self-loop term; also initializes out)
template <int F>
__global__ void gcn_self_init(const float* __restrict__ h, const float* __restrict__ dinv,
                              float* __restrict__ out, int n) {
    int i = blockIdx.x * blockDim.x + threadIdx.x;
    if (i >= n * F) return;
    const int node = i / F;   // F is a power of two -> shift
    const float di = dinv[node];
    out[i] = h[i] * di * di;
}

// out[dst,f] += h[src,f] * dinv[src] * dinv[dst]   (one thread per edge-feature)
template <int F>
__global__ void gcn_edge_agg(const int* __restrict__ src, const int* __restrict__ dst,
                             const float* __restrict__ h, const float* __restrict__ dinv,
                             float* __restrict__ out, int nE) {
    const int epb = 256 / F;  // edges per block
    const int e = blockIdx.x * epb + (int)(threadIdx.x / F);
    const int f = (int)(threadIdx.x % F);
    if (e >= nE) return;
    const int s = src[e];
    const int d = dst[e];
    const float v = h[(size_t)s * F + f] * (dinv[s] * dinv[d]);
    atomicAdd(&out[(size_t)d * F + f], v);
}

// out[i,f] = act(out[i,f] + b[f])
template <int F, bool RELU>
__global__ void gcn_bias_act(float* __restrict__ out, const float* __restrict__ b, int n) {
    int i = blockIdx.x * blockDim.x + threadIdx.x;
    if (i >= n * F) return;
    float v = out[i] + b[i % F];
    if (RELU) v = fmaxf(v, 0.0f);
    out[i] = v;
}

// ---------------------------------------------------------------------------
// Launcher
// ---------------------------------------------------------------------------
extern "C" void kernel_launch(void* const* d_in, const int* in_sizes, int n_in,
                              void* d_out, int out_size, void* d_ws, size_t ws_size,
                              hipStream_t stream) {
    const float* x  = (const float*)d_in[0];
    const int*   ei = (const int*)d_in[1];   // [2, N_EDGES]
    const float* W1 = (const float*)d_in[2];
    const float* b1 = (const float*)d_in[3];
    const float* W2 = (const float*)d_in[4];
    const float* b2 = (const float*)d_in[5];
    float* out = (float*)d_out;              // [N_NODES, 16]

    const int* src = ei;
    const int* dst = ei + N_EDGES;

    // Workspace layout (floats)
    float* ws   = (float*)d_ws;
    float* dinv = ws;                                 // N_NODES
    float* h1   = dinv + N_NODES;                     // N_NODES*64
    float* agg1 = h1 + (size_t)N_NODES * 64;          // N_NODES*64
    float* h2   = agg1 + (size_t)N_NODES * 64;        // N_NODES*16

    const int mtiles = (N_NODES + 15) / 16;

    // --- normalization ---
    gcn_deg_init<<<(N_NODES + 255) / 256, 256, 0, stream>>>(dinv, N_NODES);
    gcn_deg_count<<<(N_EDGES + 255) / 256, 256, 0, stream>>>(dst, dinv, N_EDGES);
    gcn_deg_rsqrt<<<(N_NODES + 255) / 256, 256, 0, stream>>>(dinv, N_NODES);

    // --- layer 1: h1 = x @ W1 ---
    gcn_wmma_gemm<64, 64><<<mtiles, 128, 0, stream>>>(x, W1, h1, N_NODES);

    // --- layer 1 aggregation: agg1 = D^-1/2 A D^-1/2 h1 ; then +b1, relu ---
    gcn_self_init<64><<<(N_NODES * 64 + 255) / 256, 256, 0, stream>>>(h1, dinv, agg1, N_NODES);
    gcn_edge_agg<64><<<(N_EDGES + 3) / 4, 256, 0, stream>>>(src, dst, h1, dinv, agg1, N_EDGES);
    gcn_bias_act<64, true><<<(N_NODES * 64 + 255) / 256, 256, 0, stream>>>(agg1, b1, N_NODES);

    // --- layer 2: h2 = agg1 @ W2 ---
    gcn_wmma_gemm<64, 16><<<mtiles, 32, 0, stream>>>(agg1, W2, h2, N_NODES);

    // --- layer 2 aggregation into d_out ; then +b2 ---
    gcn_self_init<16><<<(N_NODES * 16 + 255) / 256, 256, 0, stream>>>(h2, dinv, out, N_NODES);
    gcn_edge_agg<16><<<(N_EDGES + 15) / 16, 256, 0, stream>>>(src, dst, h2, dinv, out, N_EDGES);
    gcn_bias_act<16, false><<<(N_NODES * 16 + 255) / 256, 256, 0, stream>>>(out, b2, N_NODES);
}